// MatrixFactorization_4638564680527
// MI455X (gfx1250) — compile-verified
//
#include <hip/hip_runtime.h>

// M = U @ U^T - V @ V^T  (4096x4096 f32), then out[i] = M[x1[i], x2[i]].
// Lower-triangular tiled SYRK-diff with V_WMMA_F32_16X16X4_F32 (M is bitwise
// symmetric -> compute 528 of 1024 tiles), M staged in d_ws (64 MB; U+V+M
// ~192 MB = L2-resident on MI455X). Staging is double-buffered through LDS
// using CDNA5 async global->LDS copies (ASYNCcnt), overlapped with compute.

typedef __attribute__((ext_vector_type(2))) float v2f;
typedef __attribute__((ext_vector_type(4))) float v4f;
typedef __attribute__((ext_vector_type(8))) float v8f;
typedef __attribute__((__vector_size__(4 * sizeof(int)))) int i4;

#define DIM 4096
#define BT  128   // workgroup output tile (M and N)
#define KC  32    // k-chunk staged through LDS
#define PAD 36    // floats per LDS row: even + 16B-aligned + conflict-free
#define NT  256   // threads per block (8 wave32)
#define NSTAGE (DIM / KC)        // 128 k-stages per pass
#define TOTSTAGE (2 * NSTAGE)    // U pass then V pass

#if defined(__gfx1250__) && __has_builtin(__builtin_amdgcn_global_load_async_to_lds_b128)
#define HAS_ASYNC_LDS 1
#else
#define HAS_ASYNC_LDS 0
#endif

#if HAS_ASYNC_LDS
typedef __attribute__((address_space(1))) i4 gi4_t;   // global int4
typedef __attribute__((address_space(3))) i4 li4_t;   // LDS int4
#endif

static __device__ __forceinline__ void copy16_to_lds(const float* g, float* s)
{
#if HAS_ASYNC_LDS
    // L2 -> LDS DMA, no VGPR round trip; tracked by ASYNCcnt.
    __builtin_amdgcn_global_load_async_to_lds_b128(
        (gi4_t*)(g), (li4_t*)(s), /*offset=*/0, /*cpol=*/0);
#else
    *(v4f*)s = *(const v4f*)g;
#endif
}

static __device__ __forceinline__ void wait_async0()
{
#if HAS_ASYNC_LDS
#if __has_builtin(__builtin_amdgcn_s_wait_asynccnt)
    __builtin_amdgcn_s_wait_asynccnt(0);
#else
    asm volatile("s_wait_asynccnt 0" ::: "memory");
#endif
#endif
}

__global__ __launch_bounds__(NT)
void syrk_diff_wmma_kernel(const float* __restrict__ U,
                           const float* __restrict__ V,
                           float* __restrict__ M)
{
    __shared__ float As[2][BT * PAD];   // rows m0.., double buffered
    __shared__ float Bs[2][BT * PAD];   // rows n0.., double buffered

    // Map linear block id -> lower-triangular tile (bm >= bn).
    const int t = blockIdx.x;
    int bm = 0;
    while ((bm + 1) * (bm + 2) / 2 <= t) ++bm;   // <= 32 SALU iterations
    const int bn = t - bm * (bm + 1) / 2;

    const int m0 = bm * BT;
    const int n0 = bn * BT;

    const int tid   = threadIdx.x;
    const int wave  = tid >> 5;
    const int lane  = tid & 31;
    const int waveM = wave & 3;            // 4 waves x 32 rows
    const int waveN = wave >> 2;           // 2 waves x 64 cols
    const int lr    = lane & 15;           // row/col within a 16-frag
    const int kh    = (lane >> 4) << 1;    // lanes 16..31 hold K+2,K+3

    // LDS fragment base offsets (floats)
    int aoff[2], boff[4];
#pragma unroll
    for (int mi = 0; mi < 2; ++mi)
        aoff[mi] = (waveM * 32 + mi * 16 + lr) * PAD + kh;
#pragma unroll
    for (int ni = 0; ni < 4; ++ni)
        boff[ni] = (waveN * 64 + ni * 16 + lr) * PAD + kh;

    v8f acc[2][4];
#pragma unroll
    for (int mi = 0; mi < 2; ++mi)
#pragma unroll
        for (int ni = 0; ni < 4; ++ni)
            acc[mi][ni] = {};

    // Issue one stage's staging copies (this thread's 8 x 16B slots).
    auto issue_stage = [&](int st, int buf) {
        const int   pass = st >> 7;              // NSTAGE == 128
        const int   k0   = (st & (NSTAGE - 1)) * KC;
        const float* P   = pass ? V : U;
        const float* gA  = P + (size_t)m0 * DIM + k0;
        const float* gB  = P + (size_t)n0 * DIM + k0;
        float* sA = &As[buf][0];
        float* sB = &Bs[buf][0];
#pragma unroll
        for (int i = 0; i < 4; ++i) {
            const int f   = tid + i * NT;   // float4 slot 0..1023
            const int row = f >> 3;         // 0..127
            const int c   = (f & 7) << 2;   // 0..28 step 4
            copy16_to_lds(gA + (size_t)row * DIM + c, sA + row * PAD + c);
            copy16_to_lds(gB + (size_t)row * DIM + c, sB + row * PAD + c);
        }
    };

    // One pipeline step; NEG must be a compile-time literal at each call site
    // so the V-pass negation folds to plain v_xor (and to nothing on U pass).
    auto pipeline_step = [&](int st, bool NEG) {
        const int buf = st & 1;

        wait_async0();       // my async copies for stage st have landed
        __syncthreads();     // everyone's landed; other buffer free to refill

        if (st + 1 < TOTSTAGE)
            issue_stage(st + 1, buf ^ 1);   // overlaps with compute below

        const float* sA = &As[buf][0];
        const float* sB = &Bs[buf][0];

        // 8 wmma k-steps of K=4 each
#pragma unroll
        for (int kk = 0; kk < KC; kk += 4) {
            v2f a[2], b[4];
#pragma unroll
            for (int mi = 0; mi < 2; ++mi) {
                a[mi] = *(const v2f*)(sA + aoff[mi] + kk);
                if (NEG) a[mi] = -a[mi];    // V pass: D = (-A)*B + C
            }
#pragma unroll
            for (int ni = 0; ni < 4; ++ni)
                b[ni] = *(const v2f*)(sB + boff[ni] + kk);
#pragma unroll
            for (int mi = 0; mi < 2; ++mi)
#pragma unroll
                for (int ni = 0; ni < 4; ++ni)
                    acc[mi][ni] = __builtin_amdgcn_wmma_f32_16x16x4_f32(
                        false, a[mi], false, b[ni],
                        (short)0, acc[mi][ni], false, false);
        }
    };

    issue_stage(0, 0);   // prologue: fill buffer 0

    // U pass: accumulate +U U^T (no negation code at all)
#pragma unroll 1
    for (int st = 0; st < NSTAGE; ++st)
        pipeline_step(st, false);

    // V pass: accumulate -V V^T (constant-folded v_xor on A fragments)
#pragma unroll 1
    for (int st = NSTAGE; st < TOTSTAGE; ++st)
        pipeline_step(st, true);

    // C/D layout: VGPR v, lanes 0..15 -> row v, lanes 16..31 -> row v+8; col = lane&15
    const int rbase = m0 + waveM * 32 + ((lane >> 4) << 3);
    const int cbase = n0 + waveN * 64 + lr;
#pragma unroll
    for (int mi = 0; mi < 2; ++mi)
#pragma unroll
        for (int ni = 0; ni < 4; ++ni)
#pragma unroll
            for (int v = 0; v < 8; ++v) {
                const int r = rbase + mi * 16 + v;
                const int c = cbase + ni * 16;
                M[(size_t)r * DIM + c] = acc[mi][ni][v];
            }
}

__global__ __launch_bounds__(256)
void gather_sym_kernel(const int* __restrict__ x1,
                       const int* __restrict__ x2,
                       const float* __restrict__ M,
                       float* __restrict__ out, int n)
{
    const int i = blockIdx.x * blockDim.x + threadIdx.x;
    if (i < n) {
        const int a = x1[i], b = x2[i];
        const int r = a > b ? a : b;       // M is symmetric: read lower triangle
        const int c = a > b ? b : a;
        out[i] = M[(size_t)r * DIM + c];
    }
}

extern "C" void kernel_launch(void* const* d_in, const int* in_sizes, int n_in,
                              void* d_out, int out_size, void* d_ws, size_t ws_size,
                              hipStream_t stream)
{
    const float* U  = (const float*)d_in[0];
    const float* V  = (const float*)d_in[1];
    const int*   x1 = (const int*)d_in[2];
    const int*   x2 = (const int*)d_in[3];
    float* out = (float*)d_out;
    float* M   = (float*)d_ws;   // needs DIM*DIM*4 = 64 MB of scratch

    const int nt     = DIM / BT;              // 32
    const int ntiles = nt * (nt + 1) / 2;     // 528 lower-tri tiles
    syrk_diff_wmma_kernel<<<ntiles, NT, 0, stream>>>(U, V, M);

    const int n = in_sizes[2];
    gather_sym_kernel<<<(n + 255) / 256, 256, 0, stream>>>(x1, x2, M, out, n);
}